// EMM_NTM_9612136808510
// MI455X (gfx1250) — compile-verified
//
#include <hip/hip_runtime.h>
#include <hip/hip_bf16.h>

// ---------------------------------------------------------------------------
// NTM memory update on MI455X (gfx1250, wave32).
// All GEMMs use V_WMMA_F32_16X16X4_F32 (fp32 matches reference numerics).
// Shapes: B=512 H=1024 N=16384 M=128 R=4 S=3.
// ---------------------------------------------------------------------------

#define B_DIM 512
#define H_DIM 1024
#define N_DIM 16384
#define M_DIM 128
#define R_DIM 4

typedef float v2f __attribute__((ext_vector_type(2)));
typedef float v8f __attribute__((ext_vector_type(8)));

static __device__ __forceinline__ v8f wmma_f32(v2f a, v2f b, v8f c) {
  // D = A(16x4) * B(4x16) + C(16x16), fp32, wave32.
  return __builtin_amdgcn_wmma_f32_16x16x4_f32(false, a, false, b, (short)0, c,
                                               false, false);
}

static __device__ __forceinline__ float clamp01(float x) {
  return fminf(fmaxf(x, 0.0f), 1.0f);
}

// ---------------------------------------------------------------------------
// 1) k_t / e_t / a_t projections: (512x1024)@(1024x128)^T, one proj per
//    blockIdx.y. All three activations == clamp(x + bias, 0, 1).
//    grid (B/16, 3), block 128 (4 waves, each wave owns 32 output cols).
// ---------------------------------------------------------------------------
__global__ void proj_gemm_kernel(const float* __restrict__ h,
                                 const float* __restrict__ key_w,
                                 const float* __restrict__ key_b,
                                 const float* __restrict__ erase_w,
                                 const float* __restrict__ erase_b,
                                 const float* __restrict__ add_w,
                                 const float* __restrict__ add_b,
                                 float* __restrict__ k_t,
                                 float* __restrict__ e_t,
                                 float* __restrict__ a_t) {
  const int lane = threadIdx.x & 31, wave = threadIdx.x >> 5;
  const int row16 = lane & 15, half = lane >> 4;
  const float* W;
  const float* bias;
  float* out;
  if (blockIdx.y == 0)      { W = key_w;   bias = key_b;   out = k_t; }
  else if (blockIdx.y == 1) { W = erase_w; bias = erase_b; out = e_t; }
  else                      { W = add_w;   bias = add_b;   out = a_t; }

  const int b0 = blockIdx.x * 16;
  const int mbase = wave * 32;
  v8f acc0 = {0.f,0.f,0.f,0.f,0.f,0.f,0.f,0.f};
  v8f acc1 = acc0;

  const size_t arow = (size_t)(b0 + row16) * H_DIM;
  const size_t brow0 = (size_t)(mbase + row16) * H_DIM;
  const size_t brow1 = (size_t)(mbase + 16 + row16) * H_DIM;
#pragma unroll 4
  for (int k0 = 0; k0 < H_DIM; k0 += 4) {
    const int ka = k0 + half * 2;
    v2f a;  a.x  = h[arow + ka];       a.y  = h[arow + ka + 1];
    v2f bf0; bf0.x = W[brow0 + ka];    bf0.y = W[brow0 + ka + 1];
    v2f bf1; bf1.x = W[brow1 + ka];    bf1.y = W[brow1 + ka + 1];
    acc0 = wmma_f32(a, bf0, acc0);
    acc1 = wmma_f32(a, bf1, acc1);
  }
#pragma unroll
  for (int i = 0; i < 8; ++i) {
    const int r  = b0 + half * 8 + i;
    const int c0 = mbase + row16;
    const int c1 = mbase + 16 + row16;
    out[(size_t)r * M_DIM + c0] = clamp01(acc0[i] + bias[c0]);
    out[(size_t)r * M_DIM + c1] = clamp01(acc1[i] + bias[c1]);
  }
}

// ---------------------------------------------------------------------------
// 2) beta, gate, shift-softmax: 5 dot products of length 1024 per batch row.
//    grid B, block 128.
// ---------------------------------------------------------------------------
__global__ void scalars_kernel(const float* __restrict__ h,
                               const float* __restrict__ beta_w,
                               const float* __restrict__ beta_b,
                               const float* __restrict__ gate_w,
                               const float* __restrict__ gate_b,
                               const float* __restrict__ shift_w,
                               const float* __restrict__ shift_b,
                               float* __restrict__ beta,
                               float* __restrict__ g,
                               float* __restrict__ s) {
  const int b = blockIdx.x, tid = threadIdx.x;
  __shared__ float red[5][128];
  float acc[5] = {0.f, 0.f, 0.f, 0.f, 0.f};
  const float* hb = h + (size_t)b * H_DIM;
  for (int k = tid; k < H_DIM; k += 128) {
    const float hv = hb[k];
    acc[0] += hv * beta_w[k];
    acc[1] += hv * gate_w[k];
    acc[2] += hv * shift_w[k];
    acc[3] += hv * shift_w[H_DIM + k];
    acc[4] += hv * shift_w[2 * H_DIM + k];
  }
  for (int j = 0; j < 5; ++j) red[j][tid] = acc[j];
  __syncthreads();
  for (int st = 64; st > 0; st >>= 1) {
    if (tid < st)
      for (int j = 0; j < 5; ++j) red[j][tid] += red[j][tid + st];
    __syncthreads();
  }
  if (tid == 0) {
    beta[b] = fmaxf(red[0][0] + beta_b[0], 0.f);
    g[b]    = clamp01(red[1][0] + gate_b[0]);
    const float s0 = red[2][0] + shift_b[0];
    const float s1 = red[3][0] + shift_b[1];
    const float s2 = red[4][0] + shift_b[2];
    const float mx = fmaxf(s0, fmaxf(s1, s2));
    const float e0 = __expf(s0 - mx), e1 = __expf(s1 - mx), e2 = __expf(s2 - mx);
    const float inv = 1.f / (e0 + e1 + e2);
    s[b * 4 + 0] = e0 * inv;
    s[b * 4 + 1] = e1 * inv;
    s[b * 4 + 2] = e2 * inv;
  }
}

// ---------------------------------------------------------------------------
// 3) L2 norm of rows of a (rows x 128) matrix; one wave per row.
//    grid rows/8, block 256.
// ---------------------------------------------------------------------------
__global__ void rownorm_kernel(const float* __restrict__ src,
                               float* __restrict__ out, int rows) {
  const int wave = threadIdx.x >> 5, lane = threadIdx.x & 31;
  const int row = blockIdx.x * 8 + wave;
  if (row >= rows) return;
  float acc = 0.f;
  const float* p = src + (size_t)row * M_DIM;
#pragma unroll
  for (int k = lane; k < M_DIM; k += 32) { const float v = p[k]; acc += v * v; }
#pragma unroll
  for (int o = 16; o > 0; o >>= 1) acc += __shfl_xor(acc, o, 32);
  if (lane == 0) out[row] = sqrtf(acc);
}

// ---------------------------------------------------------------------------
// 4) Transpose m (N x M) -> mT (M x N) so sim GEMM B-operand is coalesced.
//    grid (N/32, M/32), block (32, 8), 32x33 LDS tile.
// ---------------------------------------------------------------------------
__global__ void transpose_kernel(const float* __restrict__ m,
                                 float* __restrict__ mT) {
  __shared__ float tile[32][33];
  const int n0 = blockIdx.x * 32, k0 = blockIdx.y * 32;
  const int tx = threadIdx.x, ty = threadIdx.y;
  for (int j = ty; j < 32; j += 8)
    tile[j][tx] = m[(size_t)(n0 + j) * M_DIM + k0 + tx];
  __syncthreads();
  for (int j = ty; j < 32; j += 8)
    mT[(size_t)(k0 + j) * N_DIM + n0 + tx] = tile[tx][j];
}

// ---------------------------------------------------------------------------
// 5) z[b,n] = beta[b] * (k_t[b,:] . m[n,:]) / (|k|*|m| + eps) via WMMA.
//    grid (B/16, N/128), block 128 (4 waves x 32 cols).
// ---------------------------------------------------------------------------
__global__ void sim_kernel(const float* __restrict__ kt,
                           const float* __restrict__ mT,
                           const float* __restrict__ beta,
                           const float* __restrict__ knorm,
                           const float* __restrict__ mnorm,
                           float* __restrict__ z) {
  const int lane = threadIdx.x & 31, wave = threadIdx.x >> 5;
  const int row16 = lane & 15, half = lane >> 4;
  const int b0 = blockIdx.x * 16;
  const int nbase = blockIdx.y * 128 + wave * 32;
  v8f acc0 = {0.f,0.f,0.f,0.f,0.f,0.f,0.f,0.f};
  v8f acc1 = acc0;
  const size_t arow = (size_t)(b0 + row16) * M_DIM;
#pragma unroll 4
  for (int k0 = 0; k0 < M_DIM; k0 += 4) {
    const int ka = k0 + half * 2;
    v2f a;  a.x = kt[arow + ka];  a.y = kt[arow + ka + 1];
    v2f bf0; bf0.x = mT[(size_t)ka * N_DIM + nbase + row16];
             bf0.y = mT[(size_t)(ka + 1) * N_DIM + nbase + row16];
    v2f bf1; bf1.x = mT[(size_t)ka * N_DIM + nbase + 16 + row16];
             bf1.y = mT[(size_t)(ka + 1) * N_DIM + nbase + 16 + row16];
    acc0 = wmma_f32(a, bf0, acc0);
    acc1 = wmma_f32(a, bf1, acc1);
  }
#pragma unroll
  for (int i = 0; i < 8; ++i) {
    const int r = b0 + half * 8 + i;
    const float bt = beta[r], kn = knorm[r];
    const int c0 = nbase + row16, c1 = nbase + 16 + row16;
    z[(size_t)r * N_DIM + c0] = bt * acc0[i] / (kn * mnorm[c0] + 1e-6f);
    z[(size_t)r * N_DIM + c1] = bt * acc1[i] / (kn * mnorm[c1] + 1e-6f);
  }
}

// ---------------------------------------------------------------------------
// 6) Per-row softmax stats over N: rowmax[b], rowinv[b] = 1/sum(exp(z-max)).
//    grid B, block 256.
// ---------------------------------------------------------------------------
__global__ void rowstats_kernel(const float* __restrict__ z,
                                float* __restrict__ rowmax,
                                float* __restrict__ rowinv) {
  const int b = blockIdx.x, tid = threadIdx.x;
  __shared__ float red[256];
  const float* zr = z + (size_t)b * N_DIM;
  float mx = -3.4e38f;
  for (int n = tid; n < N_DIM; n += 256) mx = fmaxf(mx, zr[n]);
  red[tid] = mx;
  __syncthreads();
  for (int st = 128; st > 0; st >>= 1) {
    if (tid < st) red[tid] = fmaxf(red[tid], red[tid + st]);
    __syncthreads();
  }
  mx = red[0];
  __syncthreads();
  float sm = 0.f;
  for (int n = tid; n < N_DIM; n += 256) sm += __expf(zr[n] - mx);
  red[tid] = sm;
  __syncthreads();
  for (int st = 128; st > 0; st >>= 1) {
    if (tid < st) red[tid] += red[tid + st];
    __syncthreads();
  }
  if (tid == 0) { rowmax[b] = mx; rowinv[b] = 1.f / red[0]; }
}

// ---------------------------------------------------------------------------
// 7) Weight update for ww (head 0) and wr[0..3] (heads 1..4):
//    w_g = g*w_c + (1-g)*w_old ; w~[n] = s0*w_g[n+1] + s1*w_g[n] + s2*w_g[n-1]
//    (circular). w_c = exp(z - max) * inv  (softmax folded in here).
//    grid (N/1024, B, 5), block 256; chunk + 2-halo staged in LDS.
// ---------------------------------------------------------------------------
__global__ void wupdate_kernel(const float* __restrict__ z,
                               const float* __restrict__ rowmax,
                               const float* __restrict__ rowinv,
                               const float* __restrict__ g,
                               const float* __restrict__ s,
                               const float* __restrict__ ww,
                               const float* __restrict__ wr,
                               float* __restrict__ ww_t,
                               float* __restrict__ wr_t) {
  const int b = blockIdx.y, hd = blockIdx.z;
  const int start = blockIdx.x * 1024;
  __shared__ float wg[1026];
  const float* wold = (hd == 0) ? ww + (size_t)b * N_DIM
                                : wr + ((size_t)(hd - 1) * B_DIM + b) * N_DIM;
  float* wout = (hd == 0) ? ww_t + (size_t)b * N_DIM
                          : wr_t + ((size_t)(hd - 1) * B_DIM + b) * N_DIM;
  const float gb = g[b], mx = rowmax[b], inv = rowinv[b];
  const float s0 = s[b * 4 + 0], s1 = s[b * 4 + 1], s2 = s[b * 4 + 2];
  const float* zb = z + (size_t)b * N_DIM;
  for (int i = threadIdx.x; i < 1026; i += 256) {
    const int n = (start + i - 1 + N_DIM) & (N_DIM - 1);
    const float wc = __expf(zb[n] - mx) * inv;
    wg[i] = gb * wc + (1.f - gb) * wold[n];
  }
  __syncthreads();
  for (int i = threadIdx.x; i < 1024; i += 256)
    wout[start + i] = s0 * wg[i + 2] + s1 * wg[i + 1] + s2 * wg[i];
}

// ---------------------------------------------------------------------------
// 8) m_t = m * (1 - ww^T @ e_t) + ww^T @ a_t  via dual WMMA (shared A frag).
//    grid N/16, block 128 (4 waves x 32 cols), K = B = 512.
// ---------------------------------------------------------------------------
__global__ void memwrite_kernel(const float* __restrict__ ww,
                                const float* __restrict__ et,
                                const float* __restrict__ at,
                                const float* __restrict__ m,
                                float* __restrict__ m_t) {
  const int lane = threadIdx.x & 31, wave = threadIdx.x >> 5;
  const int row16 = lane & 15, half = lane >> 4;
  const int n0 = blockIdx.x * 16;
  const int mbase = wave * 32;
  v8f ae0 = {0.f,0.f,0.f,0.f,0.f,0.f,0.f,0.f};
  v8f ae1 = ae0, aa0 = ae0, aa1 = ae0;
#pragma unroll 4
  for (int k0 = 0; k0 < B_DIM; k0 += 4) {
    const int kb = k0 + half * 2;
    v2f a;  a.x = ww[(size_t)kb * N_DIM + n0 + row16];
            a.y = ww[(size_t)(kb + 1) * N_DIM + n0 + row16];
    v2f be0; be0.x = et[(size_t)kb * M_DIM + mbase + row16];
             be0.y = et[(size_t)(kb + 1) * M_DIM + mbase + row16];
    v2f be1; be1.x = et[(size_t)kb * M_DIM + mbase + 16 + row16];
             be1.y = et[(size_t)(kb + 1) * M_DIM + mbase + 16 + row16];
    v2f ba0; ba0.x = at[(size_t)kb * M_DIM + mbase + row16];
             ba0.y = at[(size_t)(kb + 1) * M_DIM + mbase + row16];
    v2f ba1; ba1.x = at[(size_t)kb * M_DIM + mbase + 16 + row16];
             ba1.y = at[(size_t)(kb + 1) * M_DIM + mbase + 16 + row16];
    ae0 = wmma_f32(a, be0, ae0);
    ae1 = wmma_f32(a, be1, ae1);
    aa0 = wmma_f32(a, ba0, aa0);
    aa1 = wmma_f32(a, ba1, aa1);
  }
#pragma unroll
  for (int i = 0; i < 8; ++i) {
    const int n = n0 + half * 8 + i;
    const int c0 = mbase + row16, c1 = mbase + 16 + row16;
    const size_t o0 = (size_t)n * M_DIM + c0, o1 = (size_t)n * M_DIM + c1;
    m_t[o0] = m[o0] * (1.f - ae0[i]) + aa0[i];
    m_t[o1] = m[o1] * (1.f - ae1[i]) + aa1[i];
  }
}

// ---------------------------------------------------------------------------
// 9) r_t[b,r,m] = sum_n wr[r,b,n]*m[n,m]  (uses pre-update wr, pre-write m).
//    grid (B/16, R), block 128 (4 waves x 32 cols), K = N = 16384.
// ---------------------------------------------------------------------------
__global__ void read_kernel(const float* __restrict__ wr,
                            const float* __restrict__ m,
                            float* __restrict__ rt) {
  const int lane = threadIdx.x & 31, wave = threadIdx.x >> 5;
  const int row16 = lane & 15, half = lane >> 4;
  const int b0 = blockIdx.x * 16;
  const int r = blockIdx.y;
  const int mbase = wave * 32;
  const float* A = wr + (size_t)r * B_DIM * N_DIM;
  const size_t arow = (size_t)(b0 + row16) * N_DIM;
  v8f acc0 = {0.f,0.f,0.f,0.f,0.f,0.f,0.f,0.f};
  v8f acc1 = acc0;
#pragma unroll 4
  for (int k0 = 0; k0 < N_DIM; k0 += 4) {
    const int ka = k0 + half * 2;
    v2f a;  a.x = A[arow + ka];  a.y = A[arow + ka + 1];
    v2f bf0; bf0.x = m[(size_t)ka * M_DIM + mbase + row16];
             bf0.y = m[(size_t)(ka + 1) * M_DIM + mbase + row16];
    v2f bf1; bf1.x = m[(size_t)ka * M_DIM + mbase + 16 + row16];
             bf1.y = m[(size_t)(ka + 1) * M_DIM + mbase + 16 + row16];
    acc0 = wmma_f32(a, bf0, acc0);
    acc1 = wmma_f32(a, bf1, acc1);
  }
#pragma unroll
  for (int i = 0; i < 8; ++i) {
    const int br = b0 + half * 8 + i;
    const int c0 = mbase + row16, c1 = mbase + 16 + row16;
    rt[(size_t)br * (R_DIM * M_DIM) + r * M_DIM + c0] = acc0[i];
    rt[(size_t)br * (R_DIM * M_DIM) + r * M_DIM + c1] = acc1[i];
  }
}

// ---------------------------------------------------------------------------
// Host launcher.
// ---------------------------------------------------------------------------
extern "C" void kernel_launch(void* const* d_in, const int* in_sizes, int n_in,
                              void* d_out, int out_size, void* d_ws,
                              size_t ws_size, hipStream_t stream) {
  const float* h_t     = (const float*)d_in[0];
  const float* wr      = (const float*)d_in[1];
  const float* ww      = (const float*)d_in[2];
  const float* m       = (const float*)d_in[3];
  const float* key_w   = (const float*)d_in[4];
  const float* key_b   = (const float*)d_in[5];
  const float* beta_w  = (const float*)d_in[6];
  const float* beta_b  = (const float*)d_in[7];
  const float* gate_w  = (const float*)d_in[8];
  const float* gate_b  = (const float*)d_in[9];
  const float* shift_w = (const float*)d_in[10];
  const float* shift_b = (const float*)d_in[11];
  // d_in[12], d_in[13]: gamma (computed but unused by reference outputs)
  const float* erase_w = (const float*)d_in[14];
  const float* erase_b = (const float*)d_in[15];
  const float* add_w   = (const float*)d_in[16];
  const float* add_b   = (const float*)d_in[17];

  // Outputs, flat in return order: r_t, wr_t, ww_t, m_t (all fp32).
  float* out   = (float*)d_out;
  float* rt    = out;                                   // 512*4*128
  float* wr_t  = out + (size_t)B_DIM * R_DIM * M_DIM;   // + 262144
  float* ww_t  = wr_t + (size_t)R_DIM * B_DIM * N_DIM;  // + 33554432
  float* m_t   = ww_t + (size_t)B_DIM * N_DIM;          // + 8388608

  // Workspace layout (floats). Total ~10.71M floats (~43 MB).
  float* ws = (float*)d_ws;
  float* kt     = ws;                  // 65536
  float* et     = kt + 65536;          // 65536
  float* at     = et + 65536;          // 65536
  float* beta   = at + 65536;          // 512
  float* g      = beta + 512;          // 512
  float* s      = g + 512;             // 2048 (stride 4 per row)
  float* knorm  = s + 2048;            // 512
  float* mnorm  = knorm + 512;         // 16384
  float* mT     = mnorm + 16384;       // 2097152
  float* z      = mT + (size_t)M_DIM * N_DIM;  // 8388608
  float* rowmax = z + (size_t)B_DIM * N_DIM;   // 512
  float* rowinv = rowmax + 512;        // 512

  // 1) Projections (WMMA).
  proj_gemm_kernel<<<dim3(B_DIM / 16, 3), 128, 0, stream>>>(
      h_t, key_w, key_b, erase_w, erase_b, add_w, add_b, kt, et, at);
  // 2) beta / gate / shift softmax.
  scalars_kernel<<<B_DIM, 128, 0, stream>>>(h_t, beta_w, beta_b, gate_w,
                                            gate_b, shift_w, shift_b, beta, g,
                                            s);
  // 3) Row norms.
  rownorm_kernel<<<B_DIM / 8, 256, 0, stream>>>(kt, knorm, B_DIM);
  rownorm_kernel<<<N_DIM / 8, 256, 0, stream>>>(m, mnorm, N_DIM);
  // 4) Transpose m for coalesced sim B-operand.
  transpose_kernel<<<dim3(N_DIM / 32, M_DIM / 32), dim3(32, 8), 0, stream>>>(m,
                                                                             mT);
  // 5) z = beta * cosine-sim (WMMA).
  sim_kernel<<<dim3(B_DIM / 16, N_DIM / 128), 128, 0, stream>>>(
      kt, mT, beta, knorm, mnorm, z);
  // 6) Softmax stats per batch row.
  rowstats_kernel<<<B_DIM, 256, 0, stream>>>(z, rowmax, rowinv);
  // 7) Gate + circular-shift weight update: ww_t and wr_t (5 heads).
  wupdate_kernel<<<dim3(N_DIM / 1024, B_DIM, 5), 256, 0, stream>>>(
      z, rowmax, rowinv, g, s, ww, wr, ww_t, wr_t);
  // 8) Erase/add memory write (dual WMMA, shared A).
  memwrite_kernel<<<N_DIM / 16, 128, 0, stream>>>(ww, et, at, m, m_t);
  // 9) Reads against pre-write memory (WMMA, K=16384).
  read_kernel<<<dim3(B_DIM / 16, R_DIM), 128, 0, stream>>>(wr, m, rt);
}